// PointPWCMultiScaleLoss_82789789598122
// MI455X (gfx1250) — compile-verified
//
#include <hip/hip_runtime.h>
#include <math.h>

// ---------------------------------------------------------------------------
// PointPWC multi-scale scene-flow loss for MI455X (gfx1250, wave32).
//
// Distance tiles via V_WMMA_F32_16X16X4_F32:
//   A row m  = (-2x_m, -2y_m, -2z_m, 1)        (queries)
//   B col n  = ( x_n,   y_n,   z_n,  |p_n|^2 ) (candidates)
//   C[m][n]  = -2 q_m . c_n + |c_n|^2 ;  full d = C[m][n] + |q_m|^2 (added per lane)
// ---------------------------------------------------------------------------

typedef __attribute__((ext_vector_type(2))) float v2f;
typedef __attribute__((ext_vector_type(8))) float v8f;

__device__ __forceinline__ float wave_sum32(float v) {
#pragma unroll
  for (int off = 16; off > 0; off >>= 1)
    v += __shfl_xor(v, off, 32);
  return v;
}

// One wave handles 32 query rows [q0, q0+32). Lane L owns row q0+L.
// Iterates candidates in 16-wide tiles; two WMMAs produce the 32x16 distance
// tile which is staged through LDS, then each lane scans its own row and
// maintains a register-resident sorted top-K (ascending distance).
template <int K, bool QWARP, bool CWARP>
__device__ __forceinline__ void knn_core(
    const float* __restrict__ q, const float* __restrict__ qf, int Nq,
    const float* __restrict__ c, const float* __restrict__ cf, int Nc,
    int q0, float* dist, int* idx, float* qpt)
{
  __shared__ float tile[32][17];  // stride 17: conflict-free row reads
  const int lane = threadIdx.x;   // 0..31 (wave32)
  const int m15  = lane & 15;
  const bool hi  = lane >= 16;

  // This lane's query point (row q0+lane) and its squared norm.
  const int qi = q0 + lane;
  float qx = q[qi], qy = q[Nq + qi], qz = q[2 * Nq + qi];
  if (QWARP) { qx += qf[qi]; qy += qf[Nq + qi]; qz += qf[2 * Nq + qi]; }
  qpt[0] = qx; qpt[1] = qy; qpt[2] = qz;
  const float qnorm = qx * qx + qy * qy + qz * qz;

  // A fragments (16x4 f32): lanes 0-15 hold (K0,K1), lanes 16-31 hold (K2,K3)
  // for row M = lane&15.  a0 -> rows q0..q0+15, a1 -> rows q0+16..q0+31.
  v2f a0, a1;
  {
    const int r0 = q0 + m15, r1 = r0 + 16;
    float x0 = q[r0], y0 = q[Nq + r0], z0 = q[2 * Nq + r0];
    float x1 = q[r1], y1 = q[Nq + r1], z1 = q[2 * Nq + r1];
    if (QWARP) {
      x0 += qf[r0]; y0 += qf[Nq + r0]; z0 += qf[2 * Nq + r0];
      x1 += qf[r1]; y1 += qf[Nq + r1]; z1 += qf[2 * Nq + r1];
    }
    if (!hi) { a0[0] = -2.0f * x0; a0[1] = -2.0f * y0;
               a1[0] = -2.0f * x1; a1[1] = -2.0f * y1; }
    else     { a0[0] = -2.0f * z0; a0[1] = 1.0f;
               a1[0] = -2.0f * z1; a1[1] = 1.0f; }
  }

#pragma unroll
  for (int t = 0; t < K; t++) { dist[t] = 3.4e38f; idx[t] = 0; }

  for (int cb = 0; cb < Nc; cb += 16) {
    // B fragment (4x16 f32): lanes 0-15 hold (K0,K1)=(x_n,y_n),
    // lanes 16-31 hold (K2,K3)=(z_n,|p_n|^2), col N = lane&15.
    const int cn = cb + m15;
    float bx = c[cn], by = c[Nc + cn], bz = c[2 * Nc + cn];
    if (CWARP) { bx += cf[cn]; by += cf[Nc + cn]; bz += cf[2 * Nc + cn]; }
    v2f bf;
    if (!hi) { bf[0] = bx; bf[1] = by; }
    else     { bf[0] = bz; bf[1] = bx * bx + by * by + bz * bz; }

    v8f acc0 = {}, acc1 = {};
    acc0 = __builtin_amdgcn_wmma_f32_16x16x4_f32(false, a0, false, bf,
                                                 (short)0, acc0, false, false);
    acc1 = __builtin_amdgcn_wmma_f32_16x16x4_f32(false, a1, false, bf,
                                                 (short)0, acc1, false, false);

    // C layout: VGPR r, lanes 0-15 -> row r, lanes 16-31 -> row r+8; col=lane&15.
    const int rowoff = hi ? 8 : 0;
#pragma unroll
    for (int r = 0; r < 8; r++) {
      tile[rowoff + r][m15]      = acc0[r];
      tile[16 + rowoff + r][m15] = acc1[r];
    }
    __syncthreads();

    // Lane scans its own row; register top-K via unrolled insert (no scratch).
#pragma unroll
    for (int j = 0; j < 16; j++) {
      const float d = tile[lane][j] + qnorm;
      if (d < dist[K - 1]) {
        dist[K - 1] = d; idx[K - 1] = cb + j;
#pragma unroll
        for (int p = K - 1; p > 0; p--) {
          if (dist[p] < dist[p - 1]) {
            float td = dist[p]; dist[p] = dist[p - 1]; dist[p - 1] = td;
            int   ti = idx[p];  idx[p]  = idx[p - 1];  idx[p - 1]  = ti;
          }
        }
      }
    }
    __syncthreads();
  }
}

// ---- Kernel 1: self-kNN(p2, k=10) -> cv2[b][n][3] --------------------------
__global__ void __launch_bounds__(32)
k_self_cv2(const float* __restrict__ pts, float* __restrict__ cv_out, int N)
{
  const int b = blockIdx.y;
  const float* p = pts + (size_t)b * 3 * N;
  const int q0 = blockIdx.x * 32;
  float dist[10]; int idx[10]; float qp[3];
  knn_core<10, false, false>(p, nullptr, N, p, nullptr, N, q0, dist, idx, qp);

  float sx = 0.f, sy = 0.f, sz = 0.f;
#pragma unroll
  for (int j = 0; j < 10; j++) {
    const int n = idx[j];
    sx += p[n]; sy += p[N + n]; sz += p[2 * N + n];
  }
  const int qi = q0 + threadIdx.x;
  float* o = cv_out + ((size_t)b * N + qi) * 3;
  const float s = 1.0f / 9.0f;
  o[0] = (sx - 10.0f * qp[0]) * s;
  o[1] = (sy - 10.0f * qp[1]) * s;
  o[2] = (sz - 10.0f * qp[2]) * s;
}

// ---- Kernel 2: self-kNN(p1, k=10) -> cvw[b][n][3], smoothness partials -----
__global__ void __launch_bounds__(32)
k_self_warp(const float* __restrict__ p1, const float* __restrict__ fl,
            float* __restrict__ cvw_out, float* __restrict__ sm_part, int N)
{
  const int b = blockIdx.y;
  const float* p = p1 + (size_t)b * 3 * N;
  const float* f = fl + (size_t)b * 3 * N;
  const int q0 = blockIdx.x * 32;
  float dist[10]; int idx[10]; float qp[3];
  knn_core<10, false, false>(p, nullptr, N, p, nullptr, N, q0, dist, idx, qp);

  const int qi = q0 + threadIdx.x;
  const float fx = f[qi], fy = f[N + qi], fz = f[2 * N + qi];
  const float wx = qp[0] + fx, wy = qp[1] + fy, wz = qp[2] + fz;

  float sx = 0.f, sy = 0.f, sz = 0.f, sm = 0.f;
#pragma unroll
  for (int j = 0; j < 10; j++) {
    const int n = idx[j];
    sx += p[n] + f[n];
    sy += p[N + n] + f[N + n];
    sz += p[2 * N + n] + f[2 * N + n];
  }
#pragma unroll
  for (int j = 0; j < 9; j++) {  // k9 = 9 nearest (incl. self, zero term)
    const int n = idx[j];
    const float dx = f[n] - fx, dy = f[N + n] - fy, dz = f[2 * N + n] - fz;
    sm += sqrtf(dx * dx + dy * dy + dz * dz);
  }
  float* o = cvw_out + ((size_t)b * N + qi) * 3;
  const float s = 1.0f / 9.0f;
  o[0] = (sx - 10.0f * wx) * s;
  o[1] = (sy - 10.0f * wy) * s;
  o[2] = (sz - 10.0f * wz) * s;

  const float tot = wave_sum32(sm * 0.125f);
  if (threadIdx.x == 0) sm_part[blockIdx.y * gridDim.x + blockIdx.x] = tot;
}

// ---- Kernel 3: cross-kNN(warp -> p2, k=5): chamfer1 + curvature ------------
__global__ void __launch_bounds__(32)
k_cross_interp(const float* __restrict__ p1, const float* __restrict__ fl,
               const float* __restrict__ p2,
               const float* __restrict__ cv2, const float* __restrict__ cvw,
               float* __restrict__ ch1_part, float* __restrict__ curv_part,
               int N1, int N2)
{
  const int b = blockIdx.y;
  const float* p1b = p1 + (size_t)b * 3 * N1;
  const float* flb = fl + (size_t)b * 3 * N1;
  const float* p2b = p2 + (size_t)b * 3 * N2;
  const int q0 = blockIdx.x * 32;
  float dist[5]; int idx[5]; float qp[3];
  knn_core<5, true, false>(p1b, flb, N1, p2b, nullptr, N2, q0, dist, idx, qp);

  float wsum = 0.f, ix = 0.f, iy = 0.f, iz = 0.f;
#pragma unroll
  for (int j = 0; j < 5; j++) {
    const float w = 1.0f / (dist[j] + 1e-8f);
    wsum += w;
    const float* g = cv2 + ((size_t)b * N2 + idx[j]) * 3;
    ix += w * g[0]; iy += w * g[1]; iz += w * g[2];
  }
  const float inv = 1.0f / wsum;
  ix *= inv; iy *= inv; iz *= inv;

  const int qi = q0 + threadIdx.x;
  const float* cw = cvw + ((size_t)b * N1 + qi) * 3;
  const float ex = ix - cw[0], ey = iy - cw[1], ez = iz - cw[2];
  const float curv = ex * ex + ey * ey + ez * ez;

  const float ch = wave_sum32(dist[0]);
  const float cu = wave_sum32(curv);
  if (threadIdx.x == 0) {
    const int pb = blockIdx.y * gridDim.x + blockIdx.x;
    ch1_part[pb] = ch;
    curv_part[pb] = cu;
  }
}

// ---- Kernel 4: cross-min(p2 -> warp, k=1): chamfer2 ------------------------
__global__ void __launch_bounds__(32)
k_cross_min(const float* __restrict__ p2, const float* __restrict__ p1,
            const float* __restrict__ fl, float* __restrict__ ch2_part,
            int N2, int N1)
{
  const int b = blockIdx.y;
  const float* p2b = p2 + (size_t)b * 3 * N2;
  const float* p1b = p1 + (size_t)b * 3 * N1;
  const float* flb = fl + (size_t)b * 3 * N1;
  const int q0 = blockIdx.x * 32;
  float dist[1]; int idx[1]; float qp[3];
  knn_core<1, false, true>(p2b, nullptr, N2, p1b, flb, N1, q0, dist, idx, qp);

  const float ch = wave_sum32(dist[0]);
  if (threadIdx.x == 0) ch2_part[blockIdx.y * gridDim.x + blockIdx.x] = ch;
}

// ---- Final deterministic combine -------------------------------------------
struct FinArgs {
  const float* sm[4];
  const float* ch1[4];
  const float* cv[4];
  const float* ch2[4];
  int nb[4];
};

__global__ void k_finalize(FinArgs a, float* __restrict__ out)
{
  const float alpha[4] = {0.02f, 0.04f, 0.08f, 0.16f};
  float total = 0.f;
  for (int s = 0; s < 4; s++) {
    float sm = 0.f, c1 = 0.f, cv = 0.f, c2 = 0.f;
    for (int i = 0; i < a.nb[s]; i++) {
      sm += a.sm[s][i];
      c1 += a.ch1[s][i];
      cv += a.cv[s][i];
      c2 += a.ch2[s][i];
    }
    total += alpha[s] * ((c1 + c2) + sm + 0.3f * cv);
  }
  out[0] = 0.5f * total;  // mean over batch B=2
}

// ---------------------------------------------------------------------------
extern "C" void kernel_launch(void* const* d_in, const int* in_sizes, int n_in,
                              void* d_out, int out_size, void* d_ws, size_t ws_size,
                              hipStream_t stream)
{
  (void)in_sizes; (void)n_in; (void)out_size; (void)ws_size;
  const int NS[4] = {8192, 4096, 2048, 1024};

  const float* pc1[4]; const float* pc2[4]; const float* fl[4];
  for (int s = 0; s < 4; s++) {
    pc1[s] = (const float*)d_in[s];
    pc2[s] = (const float*)d_in[4 + s];
    fl[s]  = (const float*)d_in[8 + s];
  }

  float* ws = (float*)d_ws;
  size_t off = 0;
  float *cv2b[4], *cvwb[4], *smp[4], *c1p[4], *cvp[4], *c2p[4];
  for (int s = 0; s < 4; s++) { cv2b[s] = ws + off; off += (size_t)2 * 3 * NS[s]; }
  for (int s = 0; s < 4; s++) { cvwb[s] = ws + off; off += (size_t)2 * 3 * NS[s]; }
  for (int s = 0; s < 4; s++) {
    const int nb = NS[s] / 16;  // (N/32 blocks) * 2 batches
    smp[s] = ws + off; off += nb;
    c1p[s] = ws + off; off += nb;
    cvp[s] = ws + off; off += nb;
    c2p[s] = ws + off; off += nb;
  }

  for (int s = 0; s < 4; s++) {
    dim3 grid(NS[s] / 32, 2);
    k_self_cv2   <<<grid, 32, 0, stream>>>(pc2[s], cv2b[s], NS[s]);
    k_self_warp  <<<grid, 32, 0, stream>>>(pc1[s], fl[s], cvwb[s], smp[s], NS[s]);
    k_cross_interp<<<grid, 32, 0, stream>>>(pc1[s], fl[s], pc2[s], cv2b[s], cvwb[s],
                                            c1p[s], cvp[s], NS[s], NS[s]);
    k_cross_min  <<<grid, 32, 0, stream>>>(pc2[s], pc1[s], fl[s], c2p[s], NS[s], NS[s]);
  }

  FinArgs fa;
  for (int s = 0; s < 4; s++) {
    fa.sm[s] = smp[s]; fa.ch1[s] = c1p[s]; fa.cv[s] = cvp[s]; fa.ch2[s] = c2p[s];
    fa.nb[s] = NS[s] / 16;
  }
  k_finalize<<<1, 1, 0, stream>>>(fa, (float*)d_out);
}